// ContrastiveLoss_30872224924410
// MI455X (gfx1250) — compile-verified
//
#include <hip/hip_runtime.h>

// ---------------------------------------------------------------------------
// ContrastiveLoss on MI455X (gfx1250), algebraically reduced to O(N*D):
//   pos_i    = ||img_i||^2 + ||txt_i||^2 - 2 img_i.txt_i
//   rowsum_i = N*||img_i||^2 + S_txt - 2 img_i.T      (T = colsum(txt))
//   neg_i    = (rowsum_i - pos_i)/(N-1)
//   loss     = mean(relu(pos_i - neg_i + 1))
// Heavy per-row dot products computed with V_WMMA_F32_16X16X4_F32 via an
// LDS-staged 16-row tile engine (diagonal extraction of the 16x16 acc).
// ---------------------------------------------------------------------------

typedef float v2f __attribute__((ext_vector_type(2)));
typedef float v8f __attribute__((ext_vector_type(8)));

#define DDIM 1024
#define ROW_PAD 4
#define LDS_STRIDE (DDIM + ROW_PAD)   // 1028 floats per row -> bank stride 4
#define CS_BLOCKS 64                  // column-sum partial blocks

// ---- Kernel 1: partial column sums of text_embeddings ---------------------
__global__ void cl_colsum_partial(const float* __restrict__ txt,
                                  float* __restrict__ partialT, int rowsPerBlock) {
    int b = blockIdx.x, t = threadIdx.x;         // 256 threads, t -> float4 col
    const float* base = txt + (size_t)b * rowsPerBlock * DDIM;
    float4 acc = make_float4(0.f, 0.f, 0.f, 0.f);
    for (int r = 0; r < rowsPerBlock; ++r) {
        float4 v = ((const float4*)(base + (size_t)r * DDIM))[t];
        acc.x += v.x; acc.y += v.y; acc.z += v.z; acc.w += v.w;
    }
    ((float4*)(partialT + (size_t)b * DDIM))[t] = acc;
}

// ---- Kernel 2: reduce partials -> T[D] ------------------------------------
__global__ void cl_colsum_reduce(const float* __restrict__ partialT,
                                 float* __restrict__ T) {
    int c = blockIdx.x * blockDim.x + threadIdx.x;   // 4 blocks x 256 = 1024
    float s = 0.f;
    for (int b = 0; b < CS_BLOCKS; ++b) s += partialT[(size_t)b * DDIM + c];
    T[c] = s;
}

// ---- Kernel 3: WMMA tile engine -------------------------------------------
// Block = 256 threads (8 waves), owns 16 rows. Wave w owns K-slice [128w,128w+128).
// Per K=4 chunk, one ds_read_b64 per matrix serves as both A and B operand
// (A lane->(M=lane&15, K=khalf+{0,1}); B lane->(K=khalf+{0,1}, N=lane&15)).
__global__ void cl_tile_wmma(const float* __restrict__ img,
                             const float* __restrict__ txt,
                             const float* __restrict__ T,
                             float* __restrict__ imgsq, float* __restrict__ txtsq,
                             float* __restrict__ dotv,  float* __restrict__ crossT) {
    extern __shared__ float smem[];
    float* s_img  = smem;                        // 16*1028
    float* s_txt  = s_img + 16 * LDS_STRIDE;     // 16*1028
    float* s_T    = s_txt + 16 * LDS_STRIDE;     // 1024
    float* s_diag = s_T + DDIM;                  // 4*8*16 = 512

    const int tid  = threadIdx.x;
    const int wave = tid >> 5;
    const int lane = tid & 31;
    const size_t rowbase = (size_t)blockIdx.x * 16;

    // Coalesced global->LDS stage: 16 rows x 256 float4 each.
    for (int i = tid; i < 16 * 256; i += 256) {
        int r = i >> 8, c4 = i & 255;
        float4 vi = ((const float4*)(img + (rowbase + r) * DDIM))[c4];
        float4 vt = ((const float4*)(txt + (rowbase + r) * DDIM))[c4];
        *(float4*)(s_img + r * LDS_STRIDE + c4 * 4) = vi;
        *(float4*)(s_txt + r * LDS_STRIDE + c4 * 4) = vt;
    }
    ((float4*)s_T)[tid] = ((const float4*)T)[tid];   // 256 float4 = 1024 floats
    __syncthreads();

    v8f cd = {0,0,0,0,0,0,0,0};   // img.txt   (diag)
    v8f ci = {0,0,0,0,0,0,0,0};   // img.img   (diag)
    v8f ct = {0,0,0,0,0,0,0,0};   // txt.txt   (diag)
    v8f cg = {0,0,0,0,0,0,0,0};   // img.T     (diag)

    const int mrow  = lane & 15;
    const int khalf = (lane >> 4) * 2;           // lanes 0-15 -> K{0,1}; 16-31 -> K{2,3}
    const int kbase = wave * 128;

    #pragma unroll 4
    for (int kc = 0; kc < 32; ++kc) {
        int k = kbase + kc * 4 + khalf;
        v2f a = *(const v2f*)(s_img + mrow * LDS_STRIDE + k);  // ds_read_b64
        v2f b = *(const v2f*)(s_txt + mrow * LDS_STRIDE + k);  // ds_read_b64
        v2f g = *(const v2f*)(s_T + k);                        // broadcast read
        cd = __builtin_amdgcn_wmma_f32_16x16x4_f32(false, a, false, b, (short)0, cd, false, false);
        ci = __builtin_amdgcn_wmma_f32_16x16x4_f32(false, a, false, a, (short)0, ci, false, false);
        ct = __builtin_amdgcn_wmma_f32_16x16x4_f32(false, b, false, b, (short)0, ct, false, false);
        cg = __builtin_amdgcn_wmma_f32_16x16x4_f32(false, a, false, g, (short)0, cg, false, false);
    }

    // Diagonal (m,m): m<8 -> VGPR m, lane m ; m>=8 -> VGPR m-8, lane m+16.
    int r8 = -1, row = -1;
    if (lane < 8)        { r8 = lane;      row = lane;      }
    else if (lane >= 24) { r8 = lane - 24; row = lane - 16; }
    if (r8 >= 0) {
        float dd = 0.f, di = 0.f, dt = 0.f, dg = 0.f;
        switch (r8) {
            case 0: dd = cd[0]; di = ci[0]; dt = ct[0]; dg = cg[0]; break;
            case 1: dd = cd[1]; di = ci[1]; dt = ct[1]; dg = cg[1]; break;
            case 2: dd = cd[2]; di = ci[2]; dt = ct[2]; dg = cg[2]; break;
            case 3: dd = cd[3]; di = ci[3]; dt = ct[3]; dg = cg[3]; break;
            case 4: dd = cd[4]; di = ci[4]; dt = ct[4]; dg = cg[4]; break;
            case 5: dd = cd[5]; di = ci[5]; dt = ct[5]; dg = cg[5]; break;
            case 6: dd = cd[6]; di = ci[6]; dt = ct[6]; dg = cg[6]; break;
            case 7: dd = cd[7]; di = ci[7]; dt = ct[7]; dg = cg[7]; break;
        }
        s_diag[0 * 128 + wave * 16 + row] = dd;
        s_diag[1 * 128 + wave * 16 + row] = di;
        s_diag[2 * 128 + wave * 16 + row] = dt;
        s_diag[3 * 128 + wave * 16 + row] = dg;
    }
    __syncthreads();

    if (tid < 64) {   // 4 quantities x 16 rows: sum over 8 waves' K-slices
        int q = tid >> 4, rw = tid & 15;
        float s = 0.f;
        for (int w = 0; w < 8; ++w) s += s_diag[q * 128 + w * 16 + rw];
        float* outp = (q == 0) ? dotv : (q == 1) ? imgsq : (q == 2) ? txtsq : crossT;
        outp[rowbase + rw] = s;
    }
}

// ---- Kernel 4: finalize (S_txt reduce + hinge + mean), single block -------
__global__ void cl_finalize(const float* __restrict__ imgsq,
                            const float* __restrict__ txtsq,
                            const float* __restrict__ dotv,
                            const float* __restrict__ crossT,
                            float* __restrict__ out, int N) {
    __shared__ float red[256];
    int t = threadIdx.x;
    float s = 0.f;
    for (int i = t; i < N; i += 256) s += txtsq[i];
    red[t] = s; __syncthreads();
    for (int off = 128; off > 0; off >>= 1) {
        if (t < off) red[t] += red[t + off];
        __syncthreads();
    }
    float S_txt = red[0];
    __syncthreads();

    const float Nf = (float)N;
    float acc = 0.f;
    for (int i = t; i < N; i += 256) {
        float pos    = imgsq[i] + txtsq[i] - 2.0f * dotv[i];
        float rowsum = Nf * imgsq[i] + S_txt - 2.0f * crossT[i];
        float neg    = (rowsum - pos) / (Nf - 1.0f);
        float h      = pos - neg + 1.0f;   // MARGIN = 1.0
        acc += (h > 0.0f) ? h : 0.0f;
    }
    red[t] = acc; __syncthreads();
    for (int off = 128; off > 0; off >>= 1) {
        if (t < off) red[t] += red[t + off];
        __syncthreads();
    }
    if (t == 0) out[0] = red[0] / Nf;
}

// ---------------------------------------------------------------------------
extern "C" void kernel_launch(void* const* d_in, const int* in_sizes, int n_in,
                              void* d_out, int out_size, void* d_ws, size_t ws_size,
                              hipStream_t stream) {
    const float* img = (const float*)d_in[0];
    const float* txt = (const float*)d_in[1];
    float* out = (float*)d_out;
    float* ws  = (float*)d_ws;

    const int N = in_sizes[0] / DDIM;   // 4096

    // Workspace layout (floats):
    float* T        = ws;                 // [1024]
    float* imgsq    = ws + 2048;          // [N]
    float* txtsq    = imgsq + N;          // [N]
    float* dotv     = txtsq + N;          // [N]
    float* crossT   = dotv + N;           // [N]
    float* partialT = crossT + N;         // [CS_BLOCKS * 1024]

    cl_colsum_partial<<<CS_BLOCKS, 256, 0, stream>>>(txt, partialT, N / CS_BLOCKS);
    cl_colsum_reduce<<<DDIM / 256, 256, 0, stream>>>(partialT, T);

    const size_t lds_bytes = (size_t)(2 * 16 * LDS_STRIDE + DDIM + 512) * sizeof(float);
    cl_tile_wmma<<<N / 16, 256, lds_bytes, stream>>>(img, txt, T,
                                                     imgsq, txtsq, dotv, crossT);

    cl_finalize<<<1, 256, 0, stream>>>(imgsq, txtsq, dotv, crossT, out, N);
}